// AttnDecoderRNN_48077863912241
// MI455X (gfx1250) — compile-verified
//
#include <hip/hip_runtime.h>
#include <hip/hip_bf16.h>
#include <math.h>

typedef __attribute__((ext_vector_type(2))) float v2f;
typedef __attribute__((ext_vector_type(8))) float v8f;

// ---------------------------------------------------------------------------
// WMMA-based GEMV: y[m] = dot(W[m,:], x) + b[m]   (fp32, V_WMMA_F32_16X16X4_F32)
// One wave -> 16 rows. Per K-chunk of 16: each lane loads 8 contiguous floats
// of its row (two b128), every lane loads the matching 8-float x chunk, and we
// issue 4 chained WMMAs. B carries x in all 16 columns, so every D column holds
// the dot product; lane 0 holds rows 0..7 (VGPR 0..7), lane 16 rows 8..15.
// ---------------------------------------------------------------------------
template <bool RELU, bool PMAX>
__global__ void gemv_wmma(const float* __restrict__ W, const float* __restrict__ x,
                          const float* __restrict__ b, float* __restrict__ y,
                          int M, int K, float* __restrict__ pmax)
{
    __shared__ float smax[16];
    const int lane = threadIdx.x & 31;
    const int wv   = threadIdx.x >> 5;
    const int nw   = blockDim.x >> 5;
    if (PMAX) {
        if (threadIdx.x < 2 * nw) smax[threadIdx.x] = -INFINITY;
        __syncthreads();
    }
    const int rowBase = (blockIdx.x * nw + wv) * 16;       // wave-uniform
    const int h = lane >> 4;                                // lane half
    const int m = lane & 15;                                // local row
    int row  = rowBase + m;
    int rowc = row < M ? row : (M - 1);                     // clamp loads, keep EXEC full
    const float* wrow = W + (size_t)rowc * K;

    v8f c = {};
    if (rowBase < M) {                                      // wave-uniform guard
        #pragma unroll 2
        for (int k = 0; k < K; k += 16) {
            float4 w0 = *(const float4*)(wrow + k + 8 * h);
            float4 w1 = *(const float4*)(wrow + k + 8 * h + 4);
            float4 x0 = *(const float4*)(x + k + 8 * h);
            float4 x1 = *(const float4*)(x + k + 8 * h + 4);
            v2f a, bb;
            a = (v2f){w0.x, w0.y}; bb = (v2f){x0.x, x0.y};
            c = __builtin_amdgcn_wmma_f32_16x16x4_f32(false, a, false, bb, (short)0, c, false, false);
            a = (v2f){w0.z, w0.w}; bb = (v2f){x0.z, x0.w};
            c = __builtin_amdgcn_wmma_f32_16x16x4_f32(false, a, false, bb, (short)0, c, false, false);
            a = (v2f){w1.x, w1.y}; bb = (v2f){x1.x, x1.y};
            c = __builtin_amdgcn_wmma_f32_16x16x4_f32(false, a, false, bb, (short)0, c, false, false);
            a = (v2f){w1.z, w1.w}; bb = (v2f){x1.z, x1.w};
            c = __builtin_amdgcn_wmma_f32_16x16x4_f32(false, a, false, bb, (short)0, c, false, false);
        }
        if (lane == 0 || lane == 16) {                      // divergence only after last WMMA
            float lm = -INFINITY;
            #pragma unroll
            for (int r = 0; r < 8; ++r) {
                int rr = rowBase + 8 * h + r;
                if (rr < M) {
                    float v = c[r] + b[rr];
                    if (RELU) v = v > 0.0f ? v : 0.0f;
                    y[rr] = v;
                    if (PMAX) lm = fmaxf(lm, v);
                }
            }
            if (PMAX) smax[wv * 2 + h] = lm;
        }
    }
    if (PMAX) {
        __syncthreads();
        if (threadIdx.x == 0) {
            float mm = -INFINITY;
            for (int i = 0; i < 2 * nw; ++i) mm = fmaxf(mm, smax[i]);
            pmax[blockIdx.x] = mm;
        }
    }
}

// embedded = emb[idx]; attn_in = [embedded | h0]; comb_in[0:H] = embedded
__global__ void embed_concat(const int* __restrict__ idx, const float* __restrict__ emb,
                             const float* __restrict__ h0,
                             float* __restrict__ attn_in, float* __restrict__ comb_in)
{
    int i = blockIdx.x * blockDim.x + threadIdx.x;          // 0..1023
    int id = idx[0];
    float e = emb[(size_t)id * 1024 + i];
    attn_in[i] = e;
    attn_in[1024 + i] = h0[i];
    comb_in[i] = e;
}

__global__ void softmax512(const float* __restrict__ logits, float* __restrict__ w)
{
    __shared__ float red[512];
    int t = threadIdx.x;                                    // 512 threads
    float v = logits[t];
    red[t] = v; __syncthreads();
    for (int s = 256; s > 0; s >>= 1) { if (t < s) red[t] = fmaxf(red[t], red[t + s]); __syncthreads(); }
    float mx = red[0]; __syncthreads();
    float e = expf(v - mx);
    red[t] = e; __syncthreads();
    for (int s = 256; s > 0; s >>= 1) { if (t < s) red[t] += red[t + s]; __syncthreads(); }
    w[t] = e / red[0];
}

// comb_in[H + j] = sum_i aw[i] * enc[i*1024 + j]
__global__ void attn_apply(const float* __restrict__ aw, const float* __restrict__ enc,
                           float* __restrict__ dst)
{
    int j = blockIdx.x * blockDim.x + threadIdx.x;          // 0..1023
    float s = 0.f;
    for (int i = 0; i < 512; ++i) s = fmaf(aw[i], enc[i * 1024 + j], s);
    dst[j] = s;
}

__global__ void gru_elem(const float* __restrict__ gi, const float* __restrict__ gh,
                         const float* __restrict__ hprev,
                         float* __restrict__ hnew, float* __restrict__ xnext)
{
    const int H = 1024;
    int i = blockIdx.x * blockDim.x + threadIdx.x;          // 0..1023
    float r = 1.f / (1.f + expf(-(gi[i]         + gh[i])));
    float z = 1.f / (1.f + expf(-(gi[H + i]     + gh[H + i])));
    float n = tanhf(gi[2 * H + i] + r * gh[2 * H + i]);
    float h = (1.f - z) * n + z * hprev[i];
    hnew[i]  = h;
    xnext[i] = h;
}

__global__ void reduce_max(const float* __restrict__ pmax, int n, float* __restrict__ scal)
{
    __shared__ float red[256];
    int t = threadIdx.x;
    float m = -INFINITY;
    for (int i = t; i < n; i += 256) m = fmaxf(m, pmax[i]);
    red[t] = m; __syncthreads();
    for (int s = 128; s > 0; s >>= 1) { if (t < s) red[t] = fmaxf(red[t], red[t + s]); __syncthreads(); }
    if (t == 0) scal[0] = red[0];
}

__global__ void sumexp_part(const float* __restrict__ logits, int n,
                            const float* __restrict__ scal, float* __restrict__ psum)
{
    __shared__ float red[256];
    float gmax = scal[0];
    int t = threadIdx.x;
    float s = 0.f;
    for (int i = blockIdx.x * 256 + t; i < n; i += gridDim.x * 256) s += expf(logits[i] - gmax);
    red[t] = s; __syncthreads();
    for (int st = 128; st > 0; st >>= 1) { if (t < st) red[t] += red[t + st]; __syncthreads(); }
    if (t == 0) psum[blockIdx.x] = red[0];
}

__global__ void reduce_sum_log(const float* __restrict__ psum, int n, float* __restrict__ scal)
{
    __shared__ float red[256];
    int t = threadIdx.x;
    float s = 0.f;
    for (int i = t; i < n; i += 256) s += psum[i];
    red[t] = s; __syncthreads();
    for (int st = 128; st > 0; st >>= 1) { if (t < st) red[t] += red[t + st]; __syncthreads(); }
    if (t == 0) scal[1] = logf(red[0]);
}

__global__ void lsm_final(float* __restrict__ out, int n, const float* __restrict__ scal)
{
    int i = blockIdx.x * blockDim.x + threadIdx.x;
    if (i < n) out[i] = out[i] - scal[0] - scal[1];
}

extern "C" void kernel_launch(void* const* d_in, const int* in_sizes, int n_in,
                              void* d_out, int out_size, void* d_ws, size_t ws_size,
                              hipStream_t stream)
{
    const int H = 1024, V = 50257;
    const int*   _input   = (const int*)d_in[0];
    const float* hidden   = (const float*)d_in[1];
    const float* enc      = (const float*)d_in[2];
    const float* emb      = (const float*)d_in[3];
    const float* attn_W   = (const float*)d_in[4];
    const float* attn_b   = (const float*)d_in[5];
    const float* comb_W   = (const float*)d_in[6];
    const float* comb_b   = (const float*)d_in[7];
    const float* gru_W_ih = (const float*)d_in[8];
    const float* gru_W_hh = (const float*)d_in[9];
    const float* gru_b_ih = (const float*)d_in[10];
    const float* gru_b_hh = (const float*)d_in[11];
    const float* out_W    = (const float*)d_in[12];
    const float* out_b    = (const float*)d_in[13];

    float* out_logits = (float*)d_out;       // [V]
    float* out_hidden = out_logits + V;      // [2*H]
    float* out_attnw  = out_hidden + 2 * H;  // [512]

    float* ws          = (float*)d_ws;
    float* attn_in     = ws;                 // 2048
    float* attn_logits = ws + 2048;          // 512
    float* comb_in     = ws + 2560;          // 2048
    float* xbuf        = ws + 4608;          // 1024
    float* gi          = ws + 5632;          // 3072
    float* gh          = ws + 8704;          // 3072
    float* pmax        = ws + 11776;         // <=512
    float* psum        = ws + 12288;         // <=512
    float* scal        = ws + 12800;         // 2

    embed_concat<<<4, 256, 0, stream>>>(_input, emb, hidden, attn_in, comb_in);
    gemv_wmma<false, false><<<4, 256, 0, stream>>>(attn_W, attn_in, attn_b, attn_logits, 512, 2 * H, nullptr);
    softmax512<<<1, 512, 0, stream>>>(attn_logits, out_attnw);
    attn_apply<<<4, 256, 0, stream>>>(out_attnw, enc, comb_in + H);
    gemv_wmma<true, false><<<8, 256, 0, stream>>>(comb_W, comb_in, comb_b, xbuf, H, 2 * H, nullptr);

    for (int l = 0; l < 2; ++l) {
        gemv_wmma<false, false><<<24, 256, 0, stream>>>(gru_W_ih + (size_t)l * 3 * H * H, xbuf,
                                                        gru_b_ih + l * 3 * H, gi, 3 * H, H, nullptr);
        gemv_wmma<false, false><<<24, 256, 0, stream>>>(gru_W_hh + (size_t)l * 3 * H * H, hidden + l * H,
                                                        gru_b_hh + l * 3 * H, gh, 3 * H, H, nullptr);
        gru_elem<<<4, 256, 0, stream>>>(gi, gh, hidden + l * H, out_hidden + l * H, xbuf);
    }

    int nblkV = (V + 127) / 128;  // 393 blocks, 8 waves x 16 rows each
    gemv_wmma<false, true><<<nblkV, 256, 0, stream>>>(out_W, xbuf, out_b, out_logits, V, H, pmax);
    reduce_max<<<1, 256, 0, stream>>>(pmax, nblkV, scal);
    const int SB = 64;
    sumexp_part<<<SB, 256, 0, stream>>>(out_logits, V, scal, psum);
    reduce_sum_log<<<1, 256, 0, stream>>>(psum, SB, scal);
    lsm_final<<<(V + 255) / 256, 256, 0, stream>>>(out_logits, V, scal);
}